// Lamm_28561532518416
// MI455X (gfx1250) — compile-verified
//
#include <hip/hip_runtime.h>
#include <hip/hip_bf16.h>

typedef __attribute__((ext_vector_type(2))) float v2f;
typedef __attribute__((ext_vector_type(8))) float v8f;

#define N_LEVELS 5
#define N_BOXES 4096

// Level geometry: strides 8,16,32,64,128 of a 1333x800 input.
//  (h, w):        (100,167) (50,84) (25,42) (13,21) (7,11)
//  grid cells:    101*168=16968, 51*85=4335, 26*43=1118, 14*22=308, 8*12=96
//  ws layout (floats): [0..4] level sums, then grids at offsets below.
__constant__ int c_LVL_H[N_LEVELS]    = {100, 50, 25, 13, 7};
__constant__ int c_LVL_W[N_LEVELS]    = {167, 84, 42, 21, 11};
__constant__ int c_LVL_GOFF[N_LEVELS] = {16, 16984, 21319, 22437, 22745};
#define WS_TOTAL_FLOATS 22841

// ---------------------------------------------------------------------------
// 0) zero workspace + output
// ---------------------------------------------------------------------------
__global__ void k_zero(float* __restrict__ ws0, float* __restrict__ out, int n) {
    int i = blockIdx.x * blockDim.x + threadIdx.x;
    if (i < n) ws0[i] = 0.0f;
    if (i == 0) out[0] = 0.0f;
}

// ---------------------------------------------------------------------------
// 1) sum(h_level) via V_WMMA_F32_16X16X4_F32 with all-ones B.
//    Bandwidth-bound: float4 loads, wave-uniform grid-stride so EXEC is all
//    ones at every WMMA. Each WMMA accumulates 64 f32 elements into the
//    16x16 f32 C tile (each row-sum replicated across 16 columns -> /16).
// ---------------------------------------------------------------------------
__global__ void k_reduce_wmma(const float* __restrict__ p, int n4,
                              float* __restrict__ slot) {
    const int lane   = threadIdx.x & 31;
    const int wid    = (blockIdx.x * blockDim.x + threadIdx.x) >> 5;
    const int nwaves = (gridDim.x * blockDim.x) >> 5;

    v8f c = {0.f, 0.f, 0.f, 0.f, 0.f, 0.f, 0.f, 0.f};
    const v2f b_ones = {1.0f, 1.0f};
    const float4* __restrict__ p4 = (const float4*)p;

    // base is wave-uniform -> loop condition is wave-uniform -> EXEC all ones
    for (int base = wid * 32; base < n4; base += nwaves * 32) {
        int idx = base + lane;
        float4 v = make_float4(0.f, 0.f, 0.f, 0.f);
        if (idx < n4) v = p4[idx];          // per-lane predicated load only
        v2f a0 = {v.x, v.y};
        v2f a1 = {v.z, v.w};
        c = __builtin_amdgcn_wmma_f32_16x16x4_f32(
                false, a0, false, b_ones, (short)0, c, false, false);
        c = __builtin_amdgcn_wmma_f32_16x16x4_f32(
                false, a1, false, b_ones, (short)0, c, false, false);
    }

    // per-lane: 8 C rows at one column; sum of all lanes = 16 * data sum
    float s = c[0] + c[1] + c[2] + c[3] + c[4] + c[5] + c[6] + c[7];
    for (int off = 16; off; off >>= 1) s += __shfl_xor(s, off, 32);

    __shared__ float smem[8];
    int w8 = threadIdx.x >> 5;
    if (lane == 0) smem[w8] = s;
    __syncthreads();
    if (threadIdx.x == 0) {
        float t = 0.f;
        int nw = blockDim.x >> 5;
        for (int i = 0; i < nw; ++i) t += smem[i];
        atomicAdd(slot, t * (1.0f / 16.0f));
    }
}

// ---------------------------------------------------------------------------
// 2) rasterize boxes into per-level (h+1)x(w+1) difference grids
//    rintf == round-half-even == jnp.round
// ---------------------------------------------------------------------------
__global__ void k_rasterize(const float* __restrict__ gt,
                            const int* __restrict__ imx,
                            const int* __restrict__ imy,
                            float* __restrict__ ws0) {
    int level = blockIdx.y;
    int box = blockIdx.x * blockDim.x + threadIdx.x;
    if (box >= N_BOXES) return;

    int h = c_LVL_H[level], w = c_LVL_W[level];
    int W1 = w + 1;
    float sx = (float)w / (float)imx[0];
    float sy = (float)h / (float)imy[0];

    float4 b = ((const float4*)gt)[box];
    int x1 = min(max((int)rintf(b.x * sx), 0), w - 1);
    int y1 = min(max((int)rintf(b.y * sy), 0), h - 1);
    int x2 = min(max((int)rintf(b.z * sx), 0), w);
    int y2 = min(max((int)rintf(b.w * sy), 0), h);

    if ((x2 > x1) && (y2 > y1) && (x1 + x2 < w) && (y1 + y2 < h)) {
        float* g = ws0 + c_LVL_GOFF[level];
        atomicAdd(&g[y1 * W1 + x1],  1.0f);
        atomicAdd(&g[y1 * W1 + x2], -1.0f);
        atomicAdd(&g[y2 * W1 + x1], -1.0f);
        atomicAdd(&g[y2 * W1 + x2],  1.0f);
    }
}

// ---------------------------------------------------------------------------
// 3) per-level: column cumsum, fused row cumsum + count(>0.5), final loss
//    One block per level; grids are tiny (<=17K cells) and stay in L2.
//    cumsum is causal, so only the [0:h,0:w] region needs processing.
// ---------------------------------------------------------------------------
__global__ void k_scan_count(float* __restrict__ ws0, float* __restrict__ out) {
    int level = blockIdx.x;
    int h = c_LVL_H[level], w = c_LVL_W[level];
    int W1 = w + 1;
    float* g = ws0 + c_LVL_GOFF[level];

    // column-wise cumsum (axis 0)
    for (int x = threadIdx.x; x < w; x += blockDim.x) {
        float run = 0.f;
        for (int y = 0; y < h; ++y) {
            run += g[y * W1 + x];
            g[y * W1 + x] = run;
        }
    }
    __syncthreads();

    // row-wise cumsum (axis 1) fused with count of covered pixels
    int cnt = 0;
    for (int y = threadIdx.x; y < h; y += blockDim.x) {
        float run = 0.f;
        for (int x = 0; x < w; ++x) {
            run += g[y * W1 + x];
            if (run > 0.5f) ++cnt;
        }
    }

    __shared__ int sh[256];
    sh[threadIdx.x] = cnt;
    __syncthreads();
    for (int s = blockDim.x >> 1; s; s >>= 1) {
        if (threadIdx.x < (unsigned)s) sh[threadIdx.x] += sh[threadIdx.x + s];
        __syncthreads();
    }
    if (threadIdx.x == 0) {
        float tn = 64.0f * (float)h * (float)w;
        float pi = (float)sh[0] / tn;
        float d  = ws0[level] / tn - pi;
        atomicAdd(out, d * d * (1.0f / (float)N_LEVELS));
    }
}

// ---------------------------------------------------------------------------
// host launcher
// ---------------------------------------------------------------------------
extern "C" void kernel_launch(void* const* d_in, const int* in_sizes, int n_in,
                              void* d_out, int out_size, void* d_ws, size_t ws_size,
                              hipStream_t stream) {
    (void)in_sizes; (void)n_in; (void)out_size; (void)ws_size;

    static const int H[N_LEVELS] = {100, 50, 25, 13, 7};
    static const int W[N_LEVELS] = {167, 84, 42, 21, 11};

    const float* hlev[N_LEVELS];
    for (int l = 0; l < N_LEVELS; ++l) hlev[l] = (const float*)d_in[l];
    const float* gt  = (const float*)d_in[5];
    const int*   imx = (const int*)d_in[6];
    const int*   imy = (const int*)d_in[7];

    float* ws0 = (float*)d_ws;
    float* out = (float*)d_out;

    // 0) zero sums + grids + output
    k_zero<<<(WS_TOTAL_FLOATS + 255) / 256, 256, 0, stream>>>(ws0, out, WS_TOTAL_FLOATS);

    // 1) per-level WMMA reductions (bandwidth-dominant: ~5.7MB total)
    for (int l = 0; l < N_LEVELS; ++l) {
        int n  = 64 * H[l] * W[l];   // always divisible by 4
        int n4 = n >> 2;
        int blocks = (n4 + 255) / 256;
        if (blocks > 2048) blocks = 2048;
        k_reduce_wmma<<<blocks, 256, 0, stream>>>(hlev[l], n4, ws0 + l);
    }

    // 2) rasterize 4096 boxes into all 5 difference grids
    dim3 rg((N_BOXES + 255) / 256, N_LEVELS);
    k_rasterize<<<rg, 256, 0, stream>>>(gt, imx, imy, ws0);

    // 3) scans + counts + final loss
    k_scan_count<<<N_LEVELS, 256, 0, stream>>>(ws0, out);
}